// CRF_37761352466962
// MI455X (gfx1250) — compile-verified
//
#include <hip/hip_runtime.h>
#include <stdint.h>

#define B_        1024
#define T_        512
#define NTAG      50        // NT = NTAGS + 2
#define NTP       64        // padded tag dim for WMMA
#define START_TAG 48
#define STOP_TAG  49
#define NEG_INF_F (-10000.0f)
#define ASTRIDE   68        // LDS row stride (floats) to stagger banks
#define NSLOT     4         // TDM ring slots
#define SLOTF     (16 * NTAG)   // floats per logit slot

typedef float v2f __attribute__((ext_vector_type(2)));
typedef float v8f __attribute__((ext_vector_type(8)));
typedef unsigned int v4u __attribute__((ext_vector_type(4)));
typedef int v4i __attribute__((ext_vector_type(4)));
typedef int v8i __attribute__((ext_vector_type(8)));

// xor-swizzle across lanes (and_mask=0x1f, or=0, xor=m) -- stays inside 16-lane groups for m<16
#define SWZF(x, m) __int_as_float(__builtin_amdgcn_ds_swizzle(__float_as_int(x), (((m) << 10) | 0x1f)))

// ---------------- prep: ET[j][i] = exp(trans[i][j]) padded 64x64; stop row ----------------
__global__ __launch_bounds__(256) void crf_prep(const float* __restrict__ trans,
                                                float* __restrict__ ET,
                                                float* __restrict__ tstop) {
    int tid = threadIdx.x;
    for (int idx = tid; idx < NTP * NTP; idx += 256) {
        int j = idx >> 6, i = idx & 63;
        float v = 0.0f;
        if (i < NTAG && j < NTAG) v = __expf(trans[i * NTAG + j]);
        ET[idx] = v;
    }
    if (tid < NTP)
        tstop[tid] = (tid < NTAG) ? trans[STOP_TAG * NTAG + tid] : NEG_INF_F;
}

// ---------------- gold-path transition score ----------------
__global__ __launch_bounds__(256) void crf_tscore(const float* __restrict__ trans,
                                                  const int* __restrict__ labels,
                                                  const int* __restrict__ lens,
                                                  float* __restrict__ tsc) {
    int b = blockIdx.x * 256 + threadIdx.x;
    if (b >= B_) return;
    int len = lens[b];                 // 1..T
    int prev = START_TAG;
    float s = 0.0f;
    for (int t = 0; t < len; ++t) {
        int cur = labels[b * T_ + t];
        s += trans[cur * NTAG + prev];
        prev = cur;
    }
    s += trans[STOP_TAG * NTAG + prev];
    tsc[b] = s;
}

// ---------------- forward recursion: one wave32 per 16-batch tile ----------------
__global__ __launch_bounds__(32) void crf_forward(const float* __restrict__ logits,
                                                  const int* __restrict__ lens,
                                                  const float* __restrict__ ET,
                                                  const float* __restrict__ tstop,
                                                  const float* __restrict__ tsc,
                                                  float* __restrict__ out) {
    __shared__ float sLogit[NSLOT * SLOTF];   // TDM ring: 4 x (16 rows x 50 f32)
    __shared__ float sAlpha[16 * ASTRIDE];
    __shared__ float sC[16];

    const int lane = threadIdx.x;      // 0..31
    const int l16  = lane & 15;
    const int hi   = lane >> 4;        // 0 or 1
    const int tb   = blockIdx.x * 16;  // batch tile base

    // ---- TDM descriptor (uniform; SGPRs). Tensor = this tile: 50 x 16 rows, row stride T*NTAG.
    const uint32_t ldsBase = (uint32_t)(uintptr_t)(&sLogit[0]);   // low 32 bits of flat = LDS offset
    const uint64_t gBase   = (uint64_t)(uintptr_t)(logits + (size_t)tb * (T_ * NTAG));
    const v8i g1 = {(int)0x00020000,          // workgroup_mask=0, data_size=2 (4B)
                    (int)(NTAG << 16),        // tensor_dim0 = 50 (low16 in [31:16])
                    (int)(16 << 16),          // tensor_dim1 = 16
                    (int)(NTAG << 16),        // tile_dim0 = 50
                    (int)16,                  // tile_dim1 = 16, tile_dim2 = 0
                    (int)(T_ * NTAG),         // tensor_dim0_stride = 25600 (low32)
                    0, 0};
    const v4i z4 = {0, 0, 0, 0};
#if __clang_major__ >= 23
    const v8i z8 = {0, 0, 0, 0, 0, 0, 0, 0};
#endif

    auto tdm_issue = [&](int tt) {
        uint64_t ga = gBase + (uint64_t)tt * (NTAG * 4);
        v4u g0;
        g0.x = 1u;                                            // count=1
        g0.y = ldsBase + (uint32_t)(tt & (NSLOT - 1)) * (SLOTF * 4);
        g0.z = (uint32_t)ga;
        g0.w = ((uint32_t)(ga >> 32) & 0x01FFFFFFu) | 0x80000000u;  // addr[56:32] | type=2
#if __clang_major__ >= 23
        __builtin_amdgcn_tensor_load_to_lds(g0, g1, z4, z4, z8, 0);
#else
        __builtin_amdgcn_tensor_load_to_lds(g0, g1, z4, z4, 0);
#endif
    };

    // prologue: fill 3 slots ahead
    tdm_issue(0);
    tdm_issue(1);
    tdm_issue(2);

    // --- B operands (exp(trans)^T, 64x64) fully register-resident.
    // B layout for 4x16 f32: VGPR0 = rows K {0 | 2}, VGPR1 = rows K {1 | 3}, split at lane 16.
    v2f Bm[16][4];
#pragma unroll
    for (int kk = 0; kk < 16; ++kk) {
        int krow = 4 * kk + 2 * hi;
#pragma unroll
        for (int nt = 0; nt < 4; ++nt) {
            Bm[kk][nt].x = ET[krow * NTP + nt * 16 + l16];
            Bm[kk][nt].y = ET[(krow + 1) * NTP + nt * 16 + l16];
        }
    }

    // per-lane D-layout constants: VGPR r -> row m = r + 8*hi, column i = nt*16 + l16
    int lenSel[8];
#pragma unroll
    for (int r = 0; r < 8; ++r) lenSel[r] = lens[tb + r + 8 * hi];
    float tsv[4];
#pragma unroll
    for (int nt = 0; nt < 4; ++nt) tsv[nt] = tstop[nt * 16 + l16];

    // alpha registers, D-layout. init: NEG_INF everywhere, 0 at i == START_TAG (=48 -> nt=3, l16=0)
    float a[4][8];
#pragma unroll
    for (int nt = 0; nt < 4; ++nt)
#pragma unroll
        for (int r = 0; r < 8; ++r) a[nt][r] = NEG_INF_F;
    if (l16 == 0) {
#pragma unroll
        for (int r = 0; r < 8; ++r) a[3][r] = 0.0f;
    }

    const int wbase = 8 * hi * ASTRIDE + l16;       // alpha store base (D layout)
    const int abase = l16 * ASTRIDE + 2 * hi;       // A-operand load base (A layout)
    const int lbase = 8 * hi * NTAG + l16;          // logit slot read base (D layout)

    for (int t = 0; t < T_; ++t) {
        // keep the TDM pipeline full: issue tile t+3 (clamped; dead slots never consumed)
        int tt = t + 3;
        tdm_issue(tt < T_ ? tt : (T_ - 1));

        // 1. mirror alpha to LDS (addressed as alpha[m][j])
#pragma unroll
        for (int nt = 0; nt < 4; ++nt)
#pragma unroll
            for (int r = 0; r < 8; ++r)
                sAlpha[wbase + r * ASTRIDE + nt * 16] = a[nt][r];

        // 2. per-row rescale factor c[m] = max_j alpha[m][j], via register + swizzle reduce
        float cSel[8];
#pragma unroll
        for (int r = 0; r < 8; ++r) {
            float mx = fmaxf(fmaxf(a[0][r], a[1][r]), fmaxf(a[2][r], a[3][r]));
            mx = fmaxf(mx, SWZF(mx, 1));
            mx = fmaxf(mx, SWZF(mx, 2));
            mx = fmaxf(mx, SWZF(mx, 4));
            mx = fmaxf(mx, SWZF(mx, 8));
            cSel[r] = mx;
        }
        if (l16 == 0) {
#pragma unroll
            for (int r = 0; r < 8; ++r) sC[8 * hi + r] = cSel[r];
        }
        float cA = sC[l16];            // c for A-layout row m = l16

        // 3. q = exp(alpha - c) x exp(trans)^T  via 64x V_WMMA_F32_16X16X4_F32
        v8f acc[4] = {};
#pragma unroll
        for (int kk = 0; kk < 16; ++kk) {
            float a0 = sAlpha[abase + kk * 4];
            float a1 = sAlpha[abase + kk * 4 + 1];
            v2f A;
            A.x = __expf(a0 - cA);
            A.y = __expf(a1 - cA);
#pragma unroll
            for (int nt = 0; nt < 4; ++nt)
                acc[nt] = __builtin_amdgcn_wmma_f32_16x16x4_f32(
                    false, A, false, Bm[kk][nt], (short)0, acc[nt], false, false);
        }

        // wait until tile t has landed in its slot (loads complete in order; <=3 outstanding)
        __builtin_amdgcn_s_wait_tensorcnt(3);
        asm volatile("" ::: "memory");
        const float* sl = &sLogit[(t & (NSLOT - 1)) * SLOTF];

        // 4. epilogue: alpha' = logit + log(q) + c, pad cols -> -inf, length mask
#pragma unroll
        for (int nt = 0; nt < 4; ++nt) {
            int i = nt * 16 + l16;
#pragma unroll
            for (int r = 0; r < 8; ++r) {
                float q  = acc[nt][r];
                float lg = sl[lbase + r * NTAG + nt * 16];   // i>=NTAG lanes read junk, masked below
                float an = lg + __logf(q) + cSel[r];
                if (i >= NTAG) an = NEG_INF_F;
                a[nt][r] = (t < lenSel[r]) ? an : a[nt][r];
            }
        }
    }

    // final: out[b] = logsumexp_i(alpha[i] + trans[STOP,i]) - tscore[b]
#pragma unroll
    for (int r = 0; r < 8; ++r) {
        float v0 = a[0][r] + tsv[0];
        float v1 = a[1][r] + tsv[1];
        float v2 = a[2][r] + tsv[2];
        float v3 = a[3][r] + tsv[3];
        float mx = fmaxf(fmaxf(v0, v1), fmaxf(v2, v3));
        mx = fmaxf(mx, SWZF(mx, 1));
        mx = fmaxf(mx, SWZF(mx, 2));
        mx = fmaxf(mx, SWZF(mx, 4));
        mx = fmaxf(mx, SWZF(mx, 8));
        float s = __expf(v0 - mx) + __expf(v1 - mx) + __expf(v2 - mx) + __expf(v3 - mx);
        s += SWZF(s, 1);
        s += SWZF(s, 2);
        s += SWZF(s, 4);
        s += SWZF(s, 8);
        float lse = mx + __logf(s);
        if (l16 == 0) {
            int m = tb + 8 * hi + r;
            out[m] = lse - tsc[m];
        }
    }
}

extern "C" void kernel_launch(void* const* d_in, const int* in_sizes, int n_in,
                              void* d_out, int out_size, void* d_ws, size_t ws_size,
                              hipStream_t stream) {
    const float* logits = (const float*)d_in[0];
    const float* trans  = (const float*)d_in[1];
    const int*   labels = (const int*)d_in[2];
    const int*   lens   = (const int*)d_in[3];
    float* out = (float*)d_out;

    float* ET    = (float*)d_ws;       // 64*64 floats
    float* tstop = ET + NTP * NTP;     // 64 floats
    float* tsc   = tstop + NTP;        // 1024 floats

    crf_prep<<<1, 256, 0, stream>>>(trans, ET, tstop);
    crf_tscore<<<(B_ + 255) / 256, 256, 0, stream>>>(trans, labels, lens, tsc);
    crf_forward<<<B_ / 16, 32, 0, stream>>>(logits, lens, ET, tstop, tsc, out);
}